// S4DKernel_84361747628473
// MI455X (gfx1250) — compile-verified
//
#include <hip/hip_runtime.h>

typedef float v2f __attribute__((ext_vector_type(2)));
typedef float v8f __attribute__((ext_vector_type(8)));

#define D_MODEL   512
#define D_STATE   64
#define L_FFT     4096
#define NTHREADS  256
#define TWO_PI    6.28318530717958647692f
#define F32_EPS   1.1920928955078125e-7f

// One block per model-dim d. Stage A: Cauchy/Woodbury spectrum into LDS.
// Stage B: length-4096 IFFT as 64x64 four-step with V_WMMA_F32_16X16X4_F32.
__global__ __launch_bounds__(NTHREADS)
void s4_spectrum_ifft(const float* __restrict__ Lre_in, const float* __restrict__ Lim_in,
                      const float* __restrict__ Pre_in, const float* __restrict__ Pim_in,
                      const float* __restrict__ Bre_in, const float* __restrict__ Bim_in,
                      const float* __restrict__ Cre_in, const float* __restrict__ Cim_in,
                      const float* __restrict__ logdt_in,
                      float* __restrict__ out)
{
    __shared__ float Xre[L_FFT];           // spectrum, later reused for Z
    __shared__ float Xim[L_FFT];
    __shared__ float Etr[64], Eti[64];     // 64th roots of unity: exp(+2*pi*i*n/64)
    __shared__ float lamR[64], lamI[64];
    __shared__ float wpbR[64], wpbI[64], wppR[64], wppI[64];
    __shared__ float wcbR[64], wcbI[64], wcpR[64], wcpI[64];
    __shared__ float s_g2;                 // 2/dt

    const int d   = blockIdx.x;
    const int tid = threadIdx.x;

    // ---- per-row constants -------------------------------------------------
    if (tid < 64) {
        const int n = tid;
        const int idx = d * D_STATE + n;
        float x  = Lre_in[idx];
        float sp = (x > 20.0f) ? x : log1pf(expf(x));      // stable softplus
        lamR[n] = -sp;
        lamI[n] = Lim_in[idx];
        float pr = Pre_in[idx], pi = Pim_in[idx];
        float br = Bre_in[idx], bi = Bim_in[idx];
        float cr = Cre_in[idx], ci = Cim_in[idx];
        wpbR[n] = pr*br + pi*bi;  wpbI[n] = pr*bi - pi*br;  // conj(P)*B
        wppR[n] = pr*pr + pi*pi;  wppI[n] = 0.0f;           // conj(P)*P
        wcbR[n] = cr*br + ci*bi;  wcbI[n] = cr*bi - ci*br;  // conj(C)*B
        wcpR[n] = cr*pr + ci*pi;  wcpI[n] = cr*pi - ci*pr;  // conj(C)*P
        float ang = (TWO_PI / 64.0f) * (float)n;
        Etr[n] = cosf(ang);  Eti[n] = sinf(ang);
    }
    if (tid == 0) s_g2 = 2.0f * expf(-logdt_in[d]);
    __syncthreads();

    const float g2 = s_g2;

    // ---- Stage A: Cauchy kernel + Woodbury, K_hat -> LDS -------------------
    for (int l = tid; l < L_FFT; l += NTHREADS) {
        float th = (TWO_PI / (float)L_FFT) * (float)l;     // z = exp(-i*th)
        float zr = cosf(th), zi = -sinf(th);
        float dzr = 1.0f + zr, dzi = zi;                   // 1 + z
        float m2 = dzr*dzr + dzi*dzi;
        if (sqrtf(m2) < F32_EPS) { dzr = F32_EPS; dzi = 0.0f; m2 = dzr*dzr; }
        float inv = 1.0f / m2;
        float ur = 1.0f - zr, ui = -zi;                    // 1 - z
        float gr = g2 * ((ur*dzr + ui*dzi) * inv);         // g = (2/dt)(1-z)/(1+z)
        float gi = g2 * ((ui*dzr - ur*dzi) * inv);
        float tdr =  2.0f * dzr * inv;                     // 2/(1+z)
        float tdi = -2.0f * dzi * inv;

        float prbR=0.f, prbI=0.f, prpR=0.f, prpI=0.f;
        float crbR=0.f, crbI=0.f, crpR=0.f, crpI=0.f;
        #pragma unroll 8
        for (int n = 0; n < D_STATE; ++n) {
            float er = gr - lamR[n];
            float ei = gi - lamI[n];
            float r2 = 1.0f / (er*er + ei*ei);
            float Rr =  er * r2;
            float Ri = -ei * r2;                           // R = 1/(g - Lam)
            prbR += wpbR[n]*Rr - wpbI[n]*Ri;  prbI += wpbR[n]*Ri + wpbI[n]*Rr;
            prpR += wppR[n]*Rr - wppI[n]*Ri;  prpI += wppR[n]*Ri + wppI[n]*Rr;
            crbR += wcbR[n]*Rr - wcbI[n]*Ri;  crbI += wcbR[n]*Ri + wcbI[n]*Rr;
            crpR += wcpR[n]*Rr - wcpI[n]*Ri;  crpI += wcpR[n]*Ri + wcpI[n]*Rr;
        }
        // q = PRB / (1 + PRP); corr = CRP * q; K_hat = (2/(1+z)) * (CRB - corr)
        float denR = 1.0f + prpR, denI = prpI;
        float dinv = 1.0f / (denR*denR + denI*denI);
        float qR = (prbR*denR + prbI*denI) * dinv;
        float qI = (prbI*denR - prbR*denI) * dinv;
        float corR = crpR*qR - crpI*qI;
        float corI = crpR*qI + crpI*qR;
        float nR = crbR - corR, nI = crbI - corI;
        Xre[l] = tdr*nR - tdi*nI;
        Xim[l] = tdr*nI + tdi*nR;
    }
    __syncthreads();

    // ---- Stage B: four-step IFFT (64x64), WMMA f32 16x16x4 -----------------
    const int lane = tid & 31;
    const int half = lane >> 4;      // 0: lanes 0-15, 1: lanes 16-31
    const int li   = lane & 15;
    const int wave = tid >> 5;       // 8 waves, 2 output tiles each

    // GEMM1: Y[l2][k1] = sum_l1 X[l2 + 64*l1] * E[(l1*k1)&63]  (complex)
    v8f yr0 = {0,0,0,0,0,0,0,0}, yi0 = {0,0,0,0,0,0,0,0};
    v8f yr1 = {0,0,0,0,0,0,0,0}, yi1 = {0,0,0,0,0,0,0,0};
    #pragma unroll 1
    for (int t = 0; t < 2; ++t) {
        const int tile = wave * 2 + t;
        const int m0 = (tile >> 2) << 4;
        const int n0 = (tile & 3) << 4;
        v8f cr = {0,0,0,0,0,0,0,0}, ci = {0,0,0,0,0,0,0,0};
        const int row = m0 + li;         // l2 (M index for A frag)
        const int col = n0 + li;         // k1 (N index for B frag)
        #pragma unroll 4
        for (int kk = 0; kk < 16; ++kk) {
            const int k0 = kk * 4 + 2 * half;          // frag K per lane-half
            v2f ar, ai, br, bi, bin;
            ar.x = Xre[row + 64 * k0];     ar.y = Xre[row + 64 * (k0 + 1)];
            ai.x = Xim[row + 64 * k0];     ai.y = Xim[row + 64 * (k0 + 1)];
            const int i0 = (k0 * col) & 63, i1 = ((k0 + 1) * col) & 63;
            br.x = Etr[i0];  br.y = Etr[i1];
            bi.x = Eti[i0];  bi.y = Eti[i1];
            bin.x = -bi.x;   bin.y = -bi.y;
            cr = __builtin_amdgcn_wmma_f32_16x16x4_f32(false, ar, false, br,  (short)0, cr, false, false);
            cr = __builtin_amdgcn_wmma_f32_16x16x4_f32(false, ai, false, bin, (short)0, cr, false, false);
            ci = __builtin_amdgcn_wmma_f32_16x16x4_f32(false, ar, false, bi,  (short)0, ci, false, false);
            ci = __builtin_amdgcn_wmma_f32_16x16x4_f32(false, ai, false, br,  (short)0, ci, false, false);
        }
        if (t == 0) { yr0 = cr; yi0 = ci; } else { yr1 = cr; yi1 = ci; }
    }
    __syncthreads();   // all reads of X done before overwriting with Z

    // Twiddle: Z[l2][k1] = Y * exp(+2*pi*i*k1*l2/4096), stored at Xre/Xim[64*l2+k1]
    #pragma unroll 1
    for (int t = 0; t < 2; ++t) {
        const int tile = wave * 2 + t;
        const int m0 = (tile >> 2) << 4;
        const int n0 = (tile & 3) << 4;
        const int n  = n0 + li;                      // k1
        #pragma unroll
        for (int v = 0; v < 8; ++v) {
            const int m = m0 + v + 8 * half;         // l2
            float ang = (TWO_PI / (float)L_FFT) * (float)(m * n);
            float twr = cosf(ang), twi = sinf(ang);
            float a = (t == 0) ? yr0[v] : yr1[v];
            float b = (t == 0) ? yi0[v] : yi1[v];
            Xre[64 * m + n] = a * twr - b * twi;
            Xim[64 * m + n] = a * twi + b * twr;
        }
    }
    __syncthreads();

    // GEMM2 (real part only): Out[k2][k1] = sum_l2 E[(k2*l2)&63] * Z[64*l2+k1]
    #pragma unroll 1
    for (int t = 0; t < 2; ++t) {
        const int tile = wave * 2 + t;
        const int m0 = (tile >> 2) << 4;
        const int n0 = (tile & 3) << 4;
        v8f orr = {0,0,0,0,0,0,0,0};
        const int row = m0 + li;          // k2 (M index for A frag)
        const int col = n0 + li;          // k1 (N index for B frag)
        #pragma unroll 4
        for (int kk = 0; kk < 16; ++kk) {
            const int k0 = kk * 4 + 2 * half;
            v2f ar, ain, br, bi;
            const int i0 = (row * k0) & 63, i1 = (row * (k0 + 1)) & 63;
            ar.x  =  Etr[i0];  ar.y  =  Etr[i1];
            ain.x = -Eti[i0];  ain.y = -Eti[i1];
            br.x = Xre[64 * k0 + col];   br.y = Xre[64 * (k0 + 1) + col];
            bi.x = Xim[64 * k0 + col];   bi.y = Xim[64 * (k0 + 1) + col];
            orr = __builtin_amdgcn_wmma_f32_16x16x4_f32(false, ar,  false, br, (short)0, orr, false, false);
            orr = __builtin_amdgcn_wmma_f32_16x16x4_f32(false, ain, false, bi, (short)0, orr, false, false);
        }
        const int n = n0 + li;
        #pragma unroll
        for (int v = 0; v < 8; ++v) {
            const int m = m0 + v + 8 * half;        // k2
            out[(size_t)d * L_FFT + 64 * m + n] = orr[v] * (1.0f / (float)L_FFT);
        }
    }
}

__global__ void copy_D(const float* __restrict__ Din, float* __restrict__ out, int n)
{
    int i = blockIdx.x * blockDim.x + threadIdx.x;
    if (i < n) out[i] = Din[i];
}

extern "C" void kernel_launch(void* const* d_in, const int* in_sizes, int n_in,
                              void* d_out, int out_size, void* d_ws, size_t ws_size,
                              hipStream_t stream)
{
    const float* Lre   = (const float*)d_in[0];
    const float* Lim   = (const float*)d_in[1];
    const float* Pre   = (const float*)d_in[2];
    const float* Pim   = (const float*)d_in[3];
    const float* Bre   = (const float*)d_in[4];
    const float* Bim   = (const float*)d_in[5];
    const float* Cre   = (const float*)d_in[6];
    const float* Cim   = (const float*)d_in[7];
    const float* Dpass = (const float*)d_in[8];
    const float* logdt = (const float*)d_in[9];
    // d_in[10] is L (device scalar); layout is fixed at L=4096 per reference.

    float* out = (float*)d_out;

    s4_spectrum_ifft<<<dim3(D_MODEL), dim3(NTHREADS), 0, stream>>>(
        Lre, Lim, Pre, Pim, Bre, Bim, Cre, Cim, logdt, out);

    const int nD = in_sizes[8];  // CHANNELS * D_MODEL = 512
    copy_D<<<dim3((nD + 255) / 256), dim3(256), 0, stream>>>(
        Dpass, out + (size_t)D_MODEL * L_FFT, nD);
}